// Triangulation_1640677507438
// MI455X (gfx1250) — compile-verified
//
#include <hip/hip_runtime.h>

// CDNA5 / gfx1250: wave32, f32 WMMA 16x16x4.
typedef __attribute__((ext_vector_type(2))) float v2f;
typedef __attribute__((ext_vector_type(8))) float v8f;

#define B_DIM 16
#define N_DIM 2048
#define XC    16          // X padded to 16 columns (9 real + 7 zero)
#define KSPLIT 8          // split-K waves per batch in K2

// ---------------------------------------------------------------------------
// K0: build X transposed: XpadT[b][c][i], c = 0..15 (cols 9..15 zero).
// X row = [x2x1, x2y1, x2, y2x1, y2y1, y2, x1, y1, 1, 0..0]
// ---------------------------------------------------------------------------
__global__ void k_build_x(const float* __restrict__ p1,
                          const float* __restrict__ p2,
                          float* __restrict__ xT) {
  int idx = blockIdx.x * blockDim.x + threadIdx.x;
  if (idx >= B_DIM * N_DIM) return;
  int b = idx / N_DIM;
  int i = idx - b * N_DIM;
  float x1 = p1[idx * 2 + 0], y1 = p1[idx * 2 + 1];
  float x2 = p2[idx * 2 + 0], y2 = p2[idx * 2 + 1];
  float v[XC] = {x2 * x1, x2 * y1, x2, y2 * x1, y2 * y1, y2, x1, y1, 1.0f,
                 0.f, 0.f, 0.f, 0.f, 0.f, 0.f, 0.f};
  float* o = xT + (size_t)b * XC * N_DIM + i;
#pragma unroll
  for (int c = 0; c < XC; ++c) o[(size_t)c * N_DIM] = v[c];
}

// ---------------------------------------------------------------------------
// K1: Yt[b] = (mask[b] @ Xpad[b])^T  via v_wmma_f32_16x16x4_f32.
// One wave per 16-row output tile; K-loop over N in steps of 4.
// A frag (16x4): lane l -> row m=l&15, ks=2*(l>>4): b64 {mask[m][k+ks..+1]} (NT)
// B frag (4x16): lane l -> col n=l&15:            b64 {XpadT[n][k+ks..+1]}
// C/D (16x16, 8 VGPRs): lane l, vgpr v -> row v+8*(l>>4), col l&15
//   -> stored transposed: Yt[col][row] so v=0..7 are contiguous (2x b128).
// Each lane consumes one 128B mask line per 32 k-values -> prefetch at that
// rate only (one global_prefetch_b8 per unrolled body), ~4KB ahead.
// ---------------------------------------------------------------------------
__global__ void __launch_bounds__(256)
k_gemm_maskX(const float* __restrict__ mask,
             const float* __restrict__ xT,
             float* __restrict__ yT) {
  int wave = (blockIdx.x * blockDim.x + threadIdx.x) >> 5;   // global wave id
  int lane = threadIdx.x & 31;
  int b    = wave >> 7;            // 128 row-tiles per batch
  int row0 = (wave & 127) << 4;
  int m    = lane & 15;
  int half = lane >> 4;

  const float* maskRow = mask + ((size_t)b * N_DIM + row0 + m) * N_DIM + 2 * half;
  const float* xrow    = xT + ((size_t)b * XC + m) * N_DIM + 2 * half;

  v8f acc = {};
#pragma unroll 8
  for (int k = 0; k < N_DIM; k += 4) {
    if ((k & 31) == 0)                                 // once per cacheline
      __builtin_prefetch(maskRow + k + 1024, 0, 1);    // global_prefetch_b8
    v2f a = __builtin_nontemporal_load((const v2f*)(maskRow + k)); // NT stream
    v2f bb = *(const v2f*)(xrow + k);
    acc = __builtin_amdgcn_wmma_f32_16x16x4_f32(false, a, false, bb,
                                                (short)0, acc, false, false);
  }
  // T[v + 8*half][m] -> Yt[b][m][row0 + 8*half + v], v contiguous
  float* yo = yT + ((size_t)b * XC + m) * N_DIM + row0 + 8 * half;
  float4 lo = {acc[0], acc[1], acc[2], acc[3]};
  float4 hi = {acc[4], acc[5], acc[6], acc[7]};
  *(float4*)(yo + 0) = lo;
  *(float4*)(yo + 4) = hi;
}

// ---------------------------------------------------------------------------
// K2: Mpart[b][w] = Xpad[b]^T @ Y[b] over K-slice w (split-K, deterministic).
// 8 waves per batch; A'[r][i] = XpadT[r][i], B'[i][c] = Yt[c][i]:
// both fragments are single b64 loads.
// ---------------------------------------------------------------------------
__global__ void __launch_bounds__(32 * KSPLIT)
k_gemm_xty(const float* __restrict__ xT,
           const float* __restrict__ yT,
           float* __restrict__ Mpart) {
  int b    = blockIdx.x;
  int w    = threadIdx.x >> 5;
  int lane = threadIdx.x & 31;
  int m    = lane & 15;
  int half = lane >> 4;
  int kbase = w * (N_DIM / KSPLIT);

  const float* xrow = xT + ((size_t)b * XC + m) * N_DIM + kbase + 2 * half;
  const float* yrow = yT + ((size_t)b * XC + m) * N_DIM + kbase + 2 * half;

  v8f acc = {};
#pragma unroll 8
  for (int k = 0; k < N_DIM / KSPLIT; k += 4) {
    v2f a  = *(const v2f*)(xrow + k);
    v2f bb = *(const v2f*)(yrow + k);
    acc = __builtin_amdgcn_wmma_f32_16x16x4_f32(false, a, false, bb,
                                                (short)0, acc, false, false);
  }
  float* mo = Mpart + ((size_t)(b * KSPLIT + w)) * 256 + (8 * half) * 16 + m;
#pragma unroll
  for (int v = 0; v < 8; ++v) mo[v * 16] = acc[v];
}

// ---------------------------------------------------------------------------
// Symmetric Jacobi eigen-decomposition (cyclic sweeps). A: nxn (overwritten,
// diag -> eigenvalues), V: nxn eigenvector columns.
// ---------------------------------------------------------------------------
__device__ void jacobi_sym(float* A, float* V, int n, int sweeps) {
  for (int i = 0; i < n; ++i)
    for (int j = 0; j < n; ++j) V[i * n + j] = (i == j) ? 1.0f : 0.0f;
  for (int s = 0; s < sweeps; ++s) {
    for (int p = 0; p < n - 1; ++p) {
      for (int q = p + 1; q < n; ++q) {
        float apq = A[p * n + q];
        if (fabsf(apq) < 1e-30f) continue;
        float app = A[p * n + p], aqq = A[q * n + q];
        float tau = (aqq - app) / (2.0f * apq);
        float t = (tau >= 0.0f ? 1.0f : -1.0f) /
                  (fabsf(tau) + sqrtf(1.0f + tau * tau));
        float c  = 1.0f / sqrtf(1.0f + t * t);
        float sn = t * c;
        for (int k = 0; k < n; ++k) {
          float akp = A[k * n + p], akq = A[k * n + q];
          A[k * n + p] = c * akp - sn * akq;
          A[k * n + q] = sn * akp + c * akq;
        }
        for (int k = 0; k < n; ++k) {
          float apk = A[p * n + k], aqk = A[q * n + k];
          A[p * n + k] = c * apk - sn * aqk;
          A[q * n + k] = sn * apk + c * aqk;
        }
        for (int k = 0; k < n; ++k) {
          float vkp = V[k * n + p], vkq = V[k * n + q];
          V[k * n + p] = c * vkp - sn * vkq;
          V[k * n + q] = sn * vkp + c * vkq;
        }
      }
    }
  }
}

// ---------------------------------------------------------------------------
// K3: per batch (1 thread each): sum split-K partials -> M (9x9);
// F = min-eigvec of M^T M; e2 = min-eigvec of F F^T; P2 = [[e2]x F | e2].
// ---------------------------------------------------------------------------
__global__ void k_fund(const float* __restrict__ Mpart, float* __restrict__ P2out) {
  int b = threadIdx.x;
  if (b >= B_DIM) return;

  float Mb[81];
  for (int r = 0; r < 9; ++r)
    for (int c = 0; c < 9; ++c) {
      float acc = 0.0f;
      for (int w = 0; w < KSPLIT; ++w)
        acc += Mpart[(size_t)(b * KSPLIT + w) * 256 + r * 16 + c];
      Mb[r * 9 + c] = acc;
    }

  float G[81];
  for (int r = 0; r < 9; ++r)
    for (int c = 0; c < 9; ++c) {
      float acc = 0.0f;
      for (int k = 0; k < 9; ++k) acc += Mb[k * 9 + r] * Mb[k * 9 + c];
      G[r * 9 + c] = acc;
    }
  float V9[81];
  jacobi_sym(G, V9, 9, 14);
  int mi = 0; float mv = G[0];
  for (int i = 1; i < 9; ++i)
    if (G[i * 9 + i] < mv) { mv = G[i * 9 + i]; mi = i; }
  float F[9];
  for (int i = 0; i < 9; ++i) F[i] = V9[i * 9 + mi];

  float G3[9];
  for (int i = 0; i < 3; ++i)
    for (int j = 0; j < 3; ++j) {
      float acc = 0.0f;
      for (int k = 0; k < 3; ++k) acc += F[i * 3 + k] * F[j * 3 + k];
      G3[i * 3 + j] = acc;
    }
  float V3[9];
  jacobi_sym(G3, V3, 3, 12);
  int m3 = 0; float v3 = G3[0];
  for (int i = 1; i < 3; ++i)
    if (G3[i * 3 + i] < v3) { v3 = G3[i * 3 + i]; m3 = i; }
  float e0 = V3[0 * 3 + m3], e1 = V3[1 * 3 + m3], e2v = V3[2 * 3 + m3];

  // P2 = [ [e2]_x @ F | e2 ]
  float C[9] = {0.f, -e2v, e1,  e2v, 0.f, -e0,  -e1, e0, 0.f};
  float* o = P2out + b * 12;
  for (int i = 0; i < 3; ++i)
    for (int j = 0; j < 3; ++j) {
      float acc = 0.0f;
      for (int k = 0; k < 3; ++k) acc += C[i * 3 + k] * F[k * 3 + j];
      o[i * 4 + j] = acc;
    }
  o[0 * 4 + 3] = e0;
  o[1 * 4 + 3] = e1;
  o[2 * 4 + 3] = e2v;
}

// ---------------------------------------------------------------------------
// K4: triangulate each point: 4x4 nullspace via eig(A^T A), dehomogenize.
// P1 = [I | 0] hardcoded: r0 = [-1,0,x1,0], r1 = [0,-1,y1,0].
// ---------------------------------------------------------------------------
__global__ void k_tri(const float* __restrict__ p1,
                      const float* __restrict__ p2,
                      const float* __restrict__ P2,
                      float* __restrict__ out) {
  int idx = blockIdx.x * blockDim.x + threadIdx.x;
  if (idx >= B_DIM * N_DIM) return;
  int b = idx / N_DIM;

  float x1 = p1[idx * 2 + 0], y1 = p1[idx * 2 + 1];
  float x2 = p2[idx * 2 + 0], y2 = p2[idx * 2 + 1];
  float pr[12];
#pragma unroll
  for (int k = 0; k < 12; ++k) pr[k] = P2[b * 12 + k];

  float rows[4][4];
  rows[0][0] = -1.f; rows[0][1] = 0.f;  rows[0][2] = x1; rows[0][3] = 0.f;
  rows[1][0] = 0.f;  rows[1][1] = -1.f; rows[1][2] = y1; rows[1][3] = 0.f;
#pragma unroll
  for (int j = 0; j < 4; ++j) rows[2][j] = x2 * pr[8 + j] - pr[j];
#pragma unroll
  for (int j = 0; j < 4; ++j) rows[3][j] = y2 * pr[8 + j] - pr[4 + j];

  float G[16];
  for (int a = 0; a < 4; ++a)
    for (int c = 0; c < 4; ++c) {
      float acc = 0.0f;
      for (int r = 0; r < 4; ++r) acc += rows[r][a] * rows[r][c];
      G[a * 4 + c] = acc;
    }
  float V4[16];
  jacobi_sym(G, V4, 4, 10);
  int mi = 0; float mv = G[0];
  for (int i = 1; i < 4; ++i)
    if (G[i * 4 + i] < mv) { mv = G[i * 4 + i]; mi = i; }
  float h0 = V4[0 * 4 + mi], h1 = V4[1 * 4 + mi];
  float h2 = V4[2 * 4 + mi], h3 = V4[3 * 4 + mi];
  float inv = 1.0f / h3;
  out[idx * 3 + 0] = h0 * inv;
  out[idx * 3 + 1] = h1 * inv;
  out[idx * 3 + 2] = h2 * inv;
}

// ---------------------------------------------------------------------------
extern "C" void kernel_launch(void* const* d_in, const int* in_sizes, int n_in,
                              void* d_out, int out_size, void* d_ws, size_t ws_size,
                              hipStream_t stream) {
  const float* p1   = (const float*)d_in[0];
  const float* p2   = (const float*)d_in[1];
  const float* mask = (const float*)d_in[2];
  float* out = (float*)d_out;

  float* ws    = (float*)d_ws;
  float* xT    = ws;                                         // B*16*N
  float* yT    = xT + (size_t)B_DIM * XC * N_DIM;            // B*16*N
  float* Mpart = yT + (size_t)B_DIM * XC * N_DIM;            // B*KSPLIT*256
  float* P2    = Mpart + (size_t)B_DIM * KSPLIT * 256;       // B*12

  int npts = B_DIM * N_DIM;
  k_build_x<<<(npts + 255) / 256, 256, 0, stream>>>(p1, p2, xT);

  // B * (N/16) = 2048 waves, 8 waves per 256-thread block -> 256 blocks
  k_gemm_maskX<<<(B_DIM * (N_DIM / 16)) / 8, 256, 0, stream>>>(mask, xT, yT);

  k_gemm_xty<<<B_DIM, 32 * KSPLIT, 0, stream>>>(xT, yT, Mpart);
  k_fund<<<1, 32, 0, stream>>>(Mpart, P2);
  k_tri<<<(npts + 255) / 256, 256, 0, stream>>>(p1, p2, P2, out);
}